// HungarianContrastiveLoss_5446018531491
// MI455X (gfx1250) — compile-verified
//
#include <hip/hip_runtime.h>
#include <hip/hip_bf16.h>

typedef __attribute__((ext_vector_type(2))) float v2f;
typedef __attribute__((ext_vector_type(8))) float v8f;

#define NB 256        // batches
#define NK 16         // K per batch
#define ND 1024       // embedding dim
#define NROW (NB*NK)  // 4096 rows of o
#define INV_T 20.0f   // 1/0.05

#define WMMA_F32(a, b, c) \
  __builtin_amdgcn_wmma_f32_16x16x4_f32(false, (a), false, (b), (short)0, (c), false, false)

// ---------------------------------------------------------------------------
// Kernel 1: fp32 WMMA GEMM with 4-way N-blocking per wave (4 independent
// accumulator chains), online logsumexp per row, and extraction of the
// per-batch 16x16 diagonal block.
// grid = (256 row tiles, 8 column chunks), block = 128 (4 waves).
// Each wave handles 16 N-tiles (4 groups of 4); K loop = 1024 in steps of 4.
// A chunk of 64 N-tiles is entirely inside pos (chunk<4) or neg (chunk>=4).
// ---------------------------------------------------------------------------
__global__ __launch_bounds__(128) void gemm_lse_kernel(
    const float* __restrict__ outp,   // (4096, 1024) rows of o
    const float* __restrict__ pos,    // (4096, 1024)
    const float* __restrict__ neg,    // (4096, 1024)
    float* __restrict__ blocksOut,    // (256, 16, 16) raw sim diag blocks
    float* __restrict__ pm,           // (4096, 32) partial row max
    float* __restrict__ ps)           // (4096, 32) partial row expsum
{
  __shared__ float Ald[16][1028];     // padded: stride 1028 -> conflict-free
  __shared__ float Ctile[4][16][65];  // per-wave 16x64 result staging (+pad)

  const int rowTile = blockIdx.x;     // 0..255 (batch b)
  const int chunk   = blockIdx.y;     // 0..7   (1024 columns each)
  const int tid     = threadIdx.x;
  const int wave    = tid >> 5;
  const int lane    = tid & 31;
  const int half    = lane >> 4;      // 0/1
  const int m       = lane & 15;

  // Stage the 16x1024 A tile (scaled by 1/T) into LDS, coalesced.
  for (int idx = tid; idx < 16 * ND; idx += 128) {
    int r = idx >> 10, c = idx & (ND - 1);
    Ald[r][c] = outp[((size_t)rowTile * 16 + r) * ND + c] * INV_T;
  }
  __syncthreads();

  float m_run = -INFINITY;
  float s_run = 0.0f;

  const float* arow = &Ald[m][2 * half];
  const float* src  = (chunk < 4) ? pos : neg;

  for (int it = 0; it < 4; ++it) {
    const int nt0 = chunk * 64 + wave * 16 + 4 * it;  // global N-tile 0..511
    const int r0  = (nt0 & (NB - 1)) * 16;            // row within src
    const float* brow = src + (size_t)(r0 + m) * ND + 2 * half;

    v8f acc0 = {}, acc1 = {}, acc2 = {}, acc3 = {};
#pragma unroll 2
    for (int k0 = 0; k0 < ND; k0 += 4) {
      v2f a  = *(const v2f*)(arow + k0);           // shared A frag
      v2f b0 = *(const v2f*)(brow + k0);
      v2f b1 = *(const v2f*)(brow + 16 * ND + k0);
      v2f b2 = *(const v2f*)(brow + 32 * ND + k0);
      v2f b3 = *(const v2f*)(brow + 48 * ND + k0);
      acc0 = WMMA_F32(a, b0, acc0);                // 4 independent chains
      acc1 = WMMA_F32(a, b1, acc1);
      acc2 = WMMA_F32(a, b2, acc2);
      acc3 = WMMA_F32(a, b3, acc3);
    }

    // Spill tiles to LDS: element acc[g] is (M = g + 8*half, N = m).
#pragma unroll
    for (int g = 0; g < 8; ++g) {
      int r = g + 8 * half;
      Ctile[wave][r][m]      = acc0[g];
      Ctile[wave][r][16 + m] = acc1[g];
      Ctile[wave][r][32 + m] = acc2[g];
      Ctile[wave][r][48 + m] = acc3[g];
    }
    __syncthreads();

    // Online logsumexp update: lane < 16 owns row `lane` of this tile group.
    if (lane < 16) {
      float tmax = -INFINITY;
#pragma unroll
      for (int j = 0; j < 64; ++j) tmax = fmaxf(tmax, Ctile[wave][lane][j]);
      float nm = fmaxf(m_run, tmax);
      float add = 0.0f;
#pragma unroll
      for (int j = 0; j < 64; ++j) add += __expf(Ctile[wave][lane][j] - nm);
      s_run = s_run * __expf(m_run - nm) + add;
      m_run = nm;
    }

    // Diagonal block of this batch (positive half, tile index == rowTile).
    if (nt0 + 0 == rowTile) {
#pragma unroll
      for (int g = 0; g < 8; ++g)
        blocksOut[rowTile * 256 + (g + 8 * half) * 16 + m] = acc0[g];
    }
    if (nt0 + 1 == rowTile) {
#pragma unroll
      for (int g = 0; g < 8; ++g)
        blocksOut[rowTile * 256 + (g + 8 * half) * 16 + m] = acc1[g];
    }
    if (nt0 + 2 == rowTile) {
#pragma unroll
      for (int g = 0; g < 8; ++g)
        blocksOut[rowTile * 256 + (g + 8 * half) * 16 + m] = acc2[g];
    }
    if (nt0 + 3 == rowTile) {
#pragma unroll
      for (int g = 0; g < 8; ++g)
        blocksOut[rowTile * 256 + (g + 8 * half) * 16 + m] = acc3[g];
    }
    __syncthreads();
  }

  if (lane < 16) {
    int row  = rowTile * 16 + lane;
    int slot = chunk * 4 + wave;   // 32 partials per row
    pm[row * 32 + slot] = m_run;
    ps[row * 32 + slot] = s_run;
  }
}

// ---------------------------------------------------------------------------
// Kernel 2: merge 32 partial (max, sum) pairs per row into lse[row].
// ---------------------------------------------------------------------------
__global__ void lse_merge_kernel(const float* __restrict__ pm,
                                 const float* __restrict__ ps,
                                 float* __restrict__ lse) {
  int row = blockIdx.x * blockDim.x + threadIdx.x;
  if (row >= NROW) return;
  float mmax = -INFINITY;
#pragma unroll
  for (int s = 0; s < 32; ++s) mmax = fmaxf(mmax, pm[row * 32 + s]);
  float ssum = 0.0f;
#pragma unroll
  for (int s = 0; s < 32; ++s)
    ssum += ps[row * 32 + s] * __expf(pm[row * 32 + s] - mmax);
  lse[row] = mmax + logf(ssum);
}

// ---------------------------------------------------------------------------
// Kernel 3: Jonker-Volgenant Hungarian (maximize) on each 16x16 block,
// one thread per batch (double internals to match the fp64 reference).
// Row shifts (the -lse term) don't change the assignment, so we solve on the
// raw sim block and fold lse into the loss afterwards.
// ---------------------------------------------------------------------------
__global__ void hungarian_kernel(const float* __restrict__ blocks,
                                 const float* __restrict__ lse,
                                 float* __restrict__ bpart) {
  int b = blockIdx.x * blockDim.x + threadIdx.x;
  if (b >= NB) return;
  const float* Cb = blocks + b * 256;

  double cost[16][16];
  for (int i = 0; i < 16; ++i)
    for (int j = 0; j < 16; ++j) cost[i][j] = -(double)Cb[i * 16 + j];

  const double BIG = 1e300;
  double u[17], v[17];
  int p[17], way[17];
  for (int i = 0; i < 17; ++i) { u[i] = 0.0; v[i] = 0.0; p[i] = 0; way[i] = 0; }

  for (int i = 1; i <= 16; ++i) {
    p[0] = i;
    int j0 = 0;
    double minv[17];
    bool used[17];
    for (int j = 0; j < 17; ++j) { minv[j] = BIG; used[j] = false; }
    while (true) {
      used[j0] = true;
      int i0 = p[j0];
      double delta = BIG;
      int j1 = -1;
      for (int j = 1; j <= 16; ++j) {
        if (!used[j]) {
          double cur = cost[i0 - 1][j - 1] - u[i0] - v[j];
          if (cur < minv[j]) { minv[j] = cur; way[j] = j0; }
          if (minv[j] < delta) { delta = minv[j]; j1 = j; }
        }
      }
      for (int j = 0; j <= 16; ++j) {
        if (used[j]) { u[p[j]] += delta; v[j] -= delta; }
        else          minv[j] -= delta;
      }
      j0 = j1;
      if (p[j0] == 0) break;
    }
    while (j0) { int j1 = way[j0]; p[j0] = p[j1]; j0 = j1; }
  }

  float part = 0.0f;
  for (int j = 1; j <= 16; ++j) {
    int row = p[j] - 1, col = j - 1;
    part += lse[b * 16 + row] - Cb[row * 16 + col];  // -(sim - lse)
  }
  bpart[b] = part;
}

// ---------------------------------------------------------------------------
// Kernel 4: deterministic fixed-order tree reduction -> scalar mean loss.
// ---------------------------------------------------------------------------
__global__ void reduce_kernel(const float* __restrict__ bpart,
                              float* __restrict__ out) {
  __shared__ float sm[NB];
  int t = threadIdx.x;
  sm[t] = bpart[t];
  __syncthreads();
  for (int s = NB / 2; s > 0; s >>= 1) {
    if (t < s) sm[t] += sm[t + s];
    __syncthreads();
  }
  if (t == 0) out[0] = sm[0] / (float)NROW;
}

extern "C" void kernel_launch(void* const* d_in, const int* in_sizes, int n_in,
                              void* d_out, int out_size, void* d_ws, size_t ws_size,
                              hipStream_t stream) {
  const float* outp = (const float*)d_in[0];
  const float* pos  = (const float*)d_in[1];
  const float* neg  = (const float*)d_in[2];
  float* out = (float*)d_out;

  // Workspace layout (floats): blocks | pm | ps | lse | bpart
  float* blocks = (float*)d_ws;            // 256*256
  float* pm     = blocks + NB * 256;       // 4096*32
  float* ps     = pm + NROW * 32;          // 4096*32
  float* lse    = ps + NROW * 32;          // 4096
  float* bpart  = lse + NROW;              // 256

  dim3 g1(NB, 8), b1(128);
  gemm_lse_kernel<<<g1, b1, 0, stream>>>(outp, pos, neg, blocks, pm, ps);

  lse_merge_kernel<<<dim3(NROW / 256), dim3(256), 0, stream>>>(pm, ps, lse);

  hungarian_kernel<<<dim3(8), dim3(32), 0, stream>>>(blocks, lse, bpart);

  reduce_kernel<<<dim3(1), dim3(NB), 0, stream>>>(bpart, out);
}